// Ring_GNN_52192442581197
// MI455X (gfx1250) — compile-verified
//
#include <hip/hip_runtime.h>

typedef float v2f __attribute__((ext_vector_type(2)));
typedef float v8f __attribute__((ext_vector_type(8)));

#define MM 64
#define BATCH 16
#define DOUT 64

// ---------- per-(n,d) statistics: row sums, col sums, diag, sum(diag), total ----------
__global__ void rg_stats(const float* __restrict__ X, float* __restrict__ stats,
                         float* __restrict__ sc, int Din) {
  int ch = blockIdx.x;            // n*Din + d
  int i  = threadIdx.x;           // 0..63
  const float* xb = X + (size_t)ch * MM * MM;
  float r = 0.f, c = 0.f;
  for (int j = 0; j < MM; ++j) {
    r += xb[i * MM + j];
    c += xb[j * MM + i];
  }
  float dg = xb[i * MM + i];
  float* sb = stats + (size_t)ch * 3 * MM;
  sb[0 * MM + i] = r;
  sb[1 * MM + i] = c;
  sb[2 * MM + i] = dg;
  __shared__ float red[MM];
  red[i] = dg; __syncthreads();
  for (int o = 32; o > 0; o >>= 1) { if (i < o) red[i] += red[i + o]; __syncthreads(); }
  float sumdiag = red[0]; __syncthreads();
  red[i] = r; __syncthreads();
  for (int o = 32; o > 0; o >>= 1) { if (i < o) red[i] += red[i + o]; __syncthreads(); }
  if (i == 0) { sc[ch * 2 + 0] = sumdiag; sc[ch * 2 + 1] = red[0]; }
}

// ---------- fold coeffs x stats into additive terms R[n,s,i], C[n,s,j], Dg[n,s,i] ----------
__global__ void rg_addends(const float* __restrict__ stats, const float* __restrict__ sc,
                           const float* __restrict__ coeffs, const float* __restrict__ dbias,
                           float* __restrict__ Radd, float* __restrict__ Cadd,
                           float* __restrict__ Dgb, int Din) {
  int s = blockIdx.x, n = blockIdx.y, idx = blockIdx.z;
  int i = threadIdx.x;  // 0..63
  __shared__ float cc[64 * 15];
  __shared__ float ssc[128];
  for (int t = i; t < Din * 15; t += MM) {
    int d = t / 15, b = t % 15;
    cc[t] = coeffs[((size_t)(idx * Din + d) * DOUT + s) * 15 + b];
  }
  for (int t = i; t < Din * 2; t += MM)
    ssc[t] = sc[(n * Din + (t >> 1)) * 2 + (t & 1)];
  __syncthreads();
  const float im = 1.0f / 64.0f, im2 = 1.0f / 4096.0f;
  float ra = 0.f, ca = 0.f, dga = 0.f;
  for (int d = 0; d < Din; ++d) {
    const float* sb = stats + (size_t)(n * Din + d) * 3 * MM;
    float rw = sb[i], cl = sb[MM + i], dg = sb[2 * MM + i];
    float sd = ssc[2 * d], tt = ssc[2 * d + 1];
    const float* c = cc + d * 15;
    // ops 1-5 live on the diagonal; 6,7,12,14,15 depend on i; 8,9,13 depend on j
    dga += c[0] * dg + c[1] * sd * im + c[2] * rw * im + c[3] * cl * im + c[4] * tt * im2;
    ra  += c[5] * cl * im + c[6] * rw * im + c[11] * dg + c[13] * sd * im + c[14] * tt * im2;
    ca  += c[7] * cl * im + c[8] * rw * im + c[12] * dg;
  }
  if (idx < 2) dga += dbias[idx * DOUT + s];   // eye*dbias only on j==0 terms (idx 0,1)
  size_t o = ((size_t)(idx * BATCH + n) * DOUT + s) * MM + i;
  Radd[o] = ra; Cadd[o] = ca; Dgb[o] = dga;
}

// ---------- (i,j) transpose per channel ----------
__global__ void rg_transpose(const float* __restrict__ X, float* __restrict__ Xt) {
  __shared__ float tile[16][17];
  int ch = blockIdx.x;
  int i0 = blockIdx.y * 16, j0 = blockIdx.z * 16;
  const float* xb = X + (size_t)ch * MM * MM;
  float* yb = Xt + (size_t)ch * MM * MM;
  tile[threadIdx.y][threadIdx.x] = xb[(i0 + threadIdx.y) * MM + j0 + threadIdx.x];
  __syncthreads();
  yb[(j0 + threadIdx.y) * MM + i0 + threadIdx.x] = tile[threadIdx.x][threadIdx.y];
}

// ---------- main equivariant GEMM: T = c10^T X + c11^T X^T + addends ----------
// 128-thread block = 4 waves (one per s-tile), sharing LDS-staged X / X^T rows.
// each wave computes a 16(s) x 64(j) strip of row i for batch n via WMMA f32 16x16x4
__global__ void __launch_bounds__(128)
rg_equi_gemm(const float* __restrict__ X, const float* __restrict__ Xt,
             const float* __restrict__ coeffs,
             const float* __restrict__ Radd, const float* __restrict__ Cadd,
             const float* __restrict__ Dgb,
             float* __restrict__ T, int Din, int idx,
             const float* __restrict__ P, const float* __restrict__ sw,
             const float* __restrict__ abias, float* __restrict__ Y) {
  __shared__ float smX[64 * MM];    // [d][j] row i of X,  Din<=64
  __shared__ float smXt[64 * MM];   // [d][j] row i of X^T
  int tid = threadIdx.x;
  int lane = tid & 31, wv = tid >> 5;
  int hl = lane >> 4, lm = lane & 15;
  int i = blockIdx.x, n = blockIdx.y;
  int s0 = 16 * wv;

  const float* xrow  = X  + (size_t)n * Din * MM * MM + i * MM;
  const float* xtrow = Xt + (size_t)n * Din * MM * MM + i * MM;
  for (int t = tid; t < Din * MM; t += 128) {
    int d = t >> 6, j = t & 63;
    smX[t]  = xrow[(size_t)d * MM * MM + j];
    smXt[t] = xtrow[(size_t)d * MM * MM + j];
  }
  __syncthreads();

  const float* cA = coeffs + (size_t)idx * Din * DOUT * 15 + (s0 + lm) * 15; // A row M=lm
  v8f acc[4];
#pragma unroll
  for (int t = 0; t < 4; ++t)
#pragma unroll
    for (int e = 0; e < 8; ++e) acc[t][e] = 0.0f;

  for (int d0 = 0; d0 < Din; d0 += 4) {
    int dA = d0 + 2 * hl;  // A frag: VGPR0 holds K=0/K=2, VGPR1 holds K=1/K=3
    v2f a10, a11;
    a10.x = cA[(size_t)dA * (DOUT * 15) + 9];        // op10 coeff (x)
    a10.y = cA[(size_t)(dA + 1) * (DOUT * 15) + 9];
    a11.x = cA[(size_t)dA * (DOUT * 15) + 10];       // op11 coeff (x^T)
    a11.y = cA[(size_t)(dA + 1) * (DOUT * 15) + 10];
    const float* bx0 = smX  + dA * MM;
    const float* bx1 = bx0 + MM;
    const float* bt0 = smXt + dA * MM;
    const float* bt1 = bt0 + MM;
#pragma unroll
    for (int t = 0; t < 4; ++t) {
      v2f b;
      b.x = bx0[16 * t + lm]; b.y = bx1[16 * t + lm];
      acc[t] = __builtin_amdgcn_wmma_f32_16x16x4_f32(false, a10, false, b,
                                                     (short)0, acc[t], false, false);
      b.x = bt0[16 * t + lm]; b.y = bt1[16 * t + lm];
      acc[t] = __builtin_amdgcn_wmma_f32_16x16x4_f32(false, a11, false, b,
                                                     (short)0, acc[t], false, false);
    }
  }

  size_t ab = (size_t)(idx * BATCH + n) * DOUT * MM;
  const float* Rb = Radd + ab;
  const float* Cb = Cadd + ab;
  const float* Db = Dgb + ab;
  float w0 = sw ? sw[0] : 0.f, w1 = sw ? sw[1] : 0.f;
#pragma unroll
  for (int r = 0; r < 8; ++r) {
    int s = s0 + r + 8 * hl;            // C/D layout: VGPR r -> M=r (lanes<16), r+8 (lanes>=16)
    float radd = Rb[s * MM + i];
    float dgv  = Db[s * MM + i];
#pragma unroll
    for (int t = 0; t < 4; ++t) {
      int j = 16 * t + lm;
      float v = acc[t][r] + radd + Cb[s * MM + j] + (i == j ? dgv : 0.0f);
      size_t o = (((size_t)n * DOUT + s) * MM + i) * MM + j;
      if (Y) {
        float y = w0 * v + w1 * P[o] + abias[s];
        Y[o] = fmaxf(y, 0.0f);
      } else {
        T[o] = v;
      }
    }
  }
}

// ---------- batched 64x64x64: P[n,s] = T2[n,s] @ T1[n,s] (WMMA f32) ----------
// 128-thread block = 4 waves (one per i-tile); both operand tiles staged in LDS.
__global__ void __launch_bounds__(128)
rg_bmm(const float* __restrict__ A, const float* __restrict__ Bm, float* __restrict__ P) {
  __shared__ float smA[MM * MM];    // T2[n,s] tile (16 KB)
  __shared__ float smB[MM * MM];    // T1[n,s] tile (16 KB)
  int tid = threadIdx.x;
  int lane = tid & 31, wv = tid >> 5;
  int hl = lane >> 4, lm = lane & 15;
  int s = blockIdx.x, n = blockIdx.y;
  size_t base = ((size_t)n * DOUT + s) * MM * MM;
  for (int t = tid; t < MM * MM; t += 128) {
    smA[t] = A[base + t];
    smB[t] = Bm[base + t];
  }
  __syncthreads();

  int i0 = 16 * wv;
  v8f acc[4];
#pragma unroll
  for (int t = 0; t < 4; ++t)
#pragma unroll
    for (int e = 0; e < 8; ++e) acc[t][e] = 0.0f;

  for (int k0 = 0; k0 < MM; k0 += 4) {
    int kA = k0 + 2 * hl;
    v2f a = *(const v2f*)(smA + (i0 + lm) * MM + kA);   // 8B-aligned (kA even) -> ds_load_b64
#pragma unroll
    for (int t = 0; t < 4; ++t) {
      v2f b;
      b.x = smB[kA * MM + 16 * t + lm];
      b.y = smB[(kA + 1) * MM + 16 * t + lm];
      acc[t] = __builtin_amdgcn_wmma_f32_16x16x4_f32(false, a, false, b,
                                                     (short)0, acc[t], false, false);
    }
  }
#pragma unroll
  for (int r = 0; r < 8; ++r) {
    int i = i0 + r + 8 * hl;
#pragma unroll
    for (int t = 0; t < 4; ++t)
      P[base + (size_t)i * MM + 16 * t + lm] = acc[t][r];
  }
}

// ---------- feature sums over (i,j) for x (8ch), x1 (64ch), x2 (64ch) ----------
__global__ void rg_feats(const float* __restrict__ x, const float* __restrict__ x1,
                         const float* __restrict__ x2, float* __restrict__ feats) {
  int c = blockIdx.x, n = blockIdx.y, tid = threadIdx.x;
  const float* src;
  if (c < 8)       src = x  + ((size_t)n * 8  + c)        * MM * MM;
  else if (c < 72) src = x1 + ((size_t)n * 64 + (c - 8))  * MM * MM;
  else             src = x2 + ((size_t)n * 64 + (c - 72)) * MM * MM;
  float s = 0.f;
  for (int t = tid; t < MM * MM; t += 256) s += src[t];
  __shared__ float red[256];
  red[tid] = s; __syncthreads();
  for (int o = 128; o > 0; o >>= 1) { if (tid < o) red[tid] += red[tid + o]; __syncthreads(); }
  if (tid == 0) feats[n * 136 + c] = red[0];
}

// ---------- tiny MLP head ----------
__global__ void rg_mlp(const float* __restrict__ feats, const float* __restrict__ w1,
                       const float* __restrict__ b1, const float* __restrict__ w2,
                       const float* __restrict__ b2, float* __restrict__ out) {
  __shared__ float h[16 * 32];
  int tid = threadIdx.x;
  {
    int n = tid >> 5, k = tid & 31;
    float acc = b1[k];
    for (int f = 0; f < 136; ++f) acc += feats[n * 136 + f] * w1[f * 32 + k];
    h[n * 32 + k] = fmaxf(acc, 0.0f);
  }
  __syncthreads();
  if (tid < 160) {
    int n = tid / 10, c = tid % 10;
    float acc = b2[c];
    for (int k = 0; k < 32; ++k) acc += h[n * 32 + k] * w2[k * 10 + c];
    out[n * 10 + c] = acc;
  }
}

extern "C" void kernel_launch(void* const* d_in, const int* in_sizes, int n_in,
                              void* d_out, int out_size, void* d_ws, size_t ws_size,
                              hipStream_t stream) {
  const float* x         = (const float*)d_in[0];
  const float* l1_coeffs = (const float*)d_in[1];
  const float* l1_dbias  = (const float*)d_in[2];
  const float* l1_switch = (const float*)d_in[3];
  const float* l1_abias  = (const float*)d_in[4];
  const float* l2_coeffs = (const float*)d_in[5];
  const float* l2_dbias  = (const float*)d_in[6];
  const float* l2_switch = (const float*)d_in[7];
  const float* l2_abias  = (const float*)d_in[8];
  const float* w1 = (const float*)d_in[9];
  const float* b1 = (const float*)d_in[10];
  const float* w2 = (const float*)d_in[11];
  const float* b2 = (const float*)d_in[12];

  float* ws = (float*)d_ws;
  const size_t S = (size_t)BATCH * 64 * MM * MM;   // 4,194,304 floats
  float* x1    = ws;
  float* x2    = x1 + S;
  float* xt    = x2 + S;
  float* T1    = xt + S;
  float* T2    = T1 + S;
  float* P     = T2 + S;
  float* stats = P + S;                                  // 16*64*3*64
  float* sc    = stats + (size_t)BATCH * 64 * 3 * MM;    // 16*64*2
  float* Radd  = sc + (size_t)BATCH * 64 * 2;
  float* Cadd  = Radd + (size_t)3 * BATCH * DOUT * MM;
  float* Dg    = Cadd + (size_t)3 * BATCH * DOUT * MM;
  float* feats = Dg + (size_t)3 * BATCH * DOUT * MM;

  auto layer = [&](const float* Xin, int Din, const float* coeffs, const float* dbias,
                   const float* sw, const float* ab, float* Yout) {
    rg_stats<<<BATCH * Din, 64, 0, stream>>>(Xin, stats, sc, Din);
    rg_addends<<<dim3(DOUT, BATCH, 3), 64, 0, stream>>>(stats, sc, coeffs, dbias,
                                                        Radd, Cadd, Dg, Din);
    rg_transpose<<<dim3(BATCH * Din, 4, 4), dim3(16, 16), 0, stream>>>(Xin, xt);
    dim3 gg(MM, BATCH);
    // outs[1] = A2 @ A1 ; A1 gets dbias[1], A2 none (addends kernel handled that)
    rg_equi_gemm<<<gg, 128, 0, stream>>>(Xin, xt, coeffs, Radd, Cadd, Dg, T1, Din, 1,
                                         nullptr, nullptr, nullptr, nullptr);
    rg_equi_gemm<<<gg, 128, 0, stream>>>(Xin, xt, coeffs, Radd, Cadd, Dg, T2, Din, 2,
                                         nullptr, nullptr, nullptr, nullptr);
    rg_bmm<<<dim3(DOUT, BATCH), 128, 0, stream>>>(T2, T1, P);
    // outs[0] fused with final combine: relu(sw0*T0 + sw1*P + abias)
    rg_equi_gemm<<<gg, 128, 0, stream>>>(Xin, xt, coeffs, Radd, Cadd, Dg, nullptr, Din, 0,
                                         P, sw, ab, Yout);
  };

  layer(x, 8, l1_coeffs, l1_dbias, l1_switch, l1_abias, x1);
  layer(x1, 64, l2_coeffs, l2_dbias, l2_switch, l2_abias, x2);

  rg_feats<<<dim3(136, BATCH), 256, 0, stream>>>(x, x1, x2, feats);
  rg_mlp<<<1, 512, 0, stream>>>(feats, w1, b1, w2, b2, (float*)d_out);
}